// GNN_4269197492385
// MI455X (gfx1250) — compile-verified
//
#include <hip/hip_runtime.h>
#include <float.h>
#include <math.h>

#define N_NODES  50000
#define N_EDGES  800000
#define D_FEAT   96
#define NHID     128
#define N_GRAPHS 512
#define KTOT     (2 * D_FEAT)   // 192: concat(mean_nbr, x) along K

typedef __attribute__((ext_vector_type(2))) float v2f;
typedef __attribute__((ext_vector_type(8))) float v8f;

// ---------------- init workspace ----------------
__global__ void gnn_init(float* __restrict__ msg, float* __restrict__ deg,
                         float* __restrict__ pooled) {
    int i = blockIdx.x * blockDim.x + threadIdx.x;
    if (i < N_NODES * D_FEAT) msg[i] = 0.0f;
    if (i < N_NODES)          deg[i] = 0.0f;
    if (i < N_GRAPHS * NHID)  pooled[i] = -FLT_MAX;
}

// ---------------- edge scatter: one wave32 per edge ----------------
__global__ void __launch_bounds__(256)
gnn_scatter(const float* __restrict__ x, const int* __restrict__ src,
            const int* __restrict__ dst, float* __restrict__ msg,
            float* __restrict__ deg) {
    int wid  = (blockIdx.x * blockDim.x + threadIdx.x) >> 5;
    int lane = threadIdx.x & 31;
    if (wid >= N_EDGES) return;
    int s = src[wid];
    int d = dst[wid];
    const float* xs = x + (size_t)s * D_FEAT;
    float*       md = msg + (size_t)d * D_FEAT;
#pragma unroll
    for (int f = lane; f < D_FEAT; f += 32)
        atomicAdd(&md[f], xs[f]);
    if (lane == 0) atomicAdd(&deg[d], 1.0f);
}

// ---------------- fused SAGE GEMM: h = relu([mean|x] @ [Wl;Wr]^T + bl) ----------------
// Grid: ceil(N/16) blocks, 256 threads (8 waves). Wave w computes 16x16 tile
// (rows blockIdx*16.., cols w*16..) via V_WMMA_F32_16X16X4_F32, K = 192.
__global__ void __launch_bounds__(256)
gnn_sage_gemm(const float* __restrict__ msg, const float* __restrict__ deg,
              const float* __restrict__ x,
              const float* __restrict__ Wl, const float* __restrict__ Wr,
              const float* __restrict__ bl, float* __restrict__ h) {
    // padded strides: 196 % 64 == 4, 132 % 64 == 4 -> conflict-free row reads
    __shared__ float As[16 * 196];      // 12544 B : A tile [16][192]
    __shared__ float Bs[96 * 132];      // 50688 B : half of B [96][128]

    const int tid  = threadIdx.x;
    const int row0 = blockIdx.x * 16;

    // ---- stage A tile: cols 0..95 = msg/deg (mean), 96..191 = x ----
    for (int e = tid; e < 16 * KTOT; e += 256) {
        int m  = e / KTOT;
        int kk = e - m * KTOT;
        int node = row0 + m;
        float v = 0.0f;
        if (node < N_NODES) {
            if (kk < D_FEAT) {
                float dg  = deg[node];
                float inv = 1.0f / fmaxf(dg, 1.0f);
                v = msg[(size_t)node * D_FEAT + kk] * inv;
            } else {
                v = x[(size_t)node * D_FEAT + (kk - D_FEAT)];
            }
        }
        As[m * 196 + kk] = v;
    }

    const int lane = tid & 31;
    const int wave = tid >> 5;
    const int n0   = wave * 16;         // 8 waves cover all 128 cols
    const int l16  = lane & 15;
    const int kh   = (lane >> 4) * 2;   // 0 for lanes 0-15, 2 for 16-31

    v8f acc = {};

    for (int kb = 0; kb < KTOT; kb += 96) {
        __syncthreads();                // A ready / previous B reads done
        // ---- stage B half: Bs[kloc][n] = W_sel[n][kloc] (coalesced reads of W) ----
        const float* W = (kb == 0) ? Wl : Wr;
        for (int e = tid; e < 96 * NHID; e += 256) {
            int n    = e / 96;
            int kloc = e - n * 96;
            Bs[kloc * 132 + n] = W[n * 96 + kloc];
        }
        __syncthreads();                // B ready

#pragma unroll
        for (int k = 0; k < 96; k += 4) {
            v2f a, b;
            a.x = As[l16 * 196 + kb + k + kh];
            a.y = As[l16 * 196 + kb + k + kh + 1];
            b.x = Bs[(k + kh) * 132 + n0 + l16];
            b.y = Bs[(k + kh + 1) * 132 + n0 + l16];
            acc = __builtin_amdgcn_wmma_f32_16x16x4_f32(
                false, a, false, b, (short)0, acc, false, false);
        }
    }

    // ---- epilogue: bias + relu + store (D layout: VGPR i -> row i / i+8) ----
    const int c    = n0 + l16;
    const int rsel = (lane >> 4) * 8;
    const float bias = bl[c];
#pragma unroll
    for (int i = 0; i < 8; ++i) {
        int r = row0 + i + rsel;
        if (r < N_NODES)
            h[(size_t)r * NHID + c] = fmaxf(acc[i] + bias, 0.0f);
    }
}

// ---------------- segment_max pool (h >= 0 -> int-bits atomicMax is exact) ----------------
__global__ void __launch_bounds__(256)
gnn_pool(const float* __restrict__ h, const int* __restrict__ batch,
         float* __restrict__ pooled) {
    long i = (long)blockIdx.x * blockDim.x + threadIdx.x;
    if (i >= (long)N_NODES * NHID) return;
    int node = (int)(i / NHID);
    int f    = (int)(i - (long)node * NHID);
    int g    = batch[node];
    atomicMax((int*)&pooled[g * NHID + f], __float_as_int(h[i]));
}

// ---------------- first = searchsorted(batch, arange(G), 'left') ----------------
__global__ void gnn_first(const int* __restrict__ batch, int* __restrict__ first) {
    int g = blockIdx.x * blockDim.x + threadIdx.x;
    if (g >= N_GRAPHS) return;
    int lo = 0, hi = N_NODES;
    while (lo < hi) {
        int mid = (lo + hi) >> 1;
        if (batch[mid] < g) lo = mid + 1; else hi = mid;
    }
    first[g] = lo;
}

// ---------------- head MLP + log_softmax, one block per graph ----------------
__global__ void __launch_bounds__(128)
gnn_head(const float* __restrict__ x, const int* __restrict__ first,
         const float* __restrict__ pooled,
         const float* __restrict__ W0, const float* __restrict__ b0,
         const float* __restrict__ W1, const float* __restrict__ b1,
         const float* __restrict__ W2, const float* __restrict__ b2,
         float* __restrict__ out) {
    __shared__ float xf[D_FEAT];
    __shared__ float z[2 * NHID];
    __shared__ float h2[NHID];
    const int g = blockIdx.x;
    const int t = threadIdx.x;

    int fn = first[g];
    if (t < D_FEAT) xf[t] = (fn < N_NODES) ? x[(size_t)fn * D_FEAT + t] : 0.0f;
    z[t] = pooled[g * NHID + t];
    __syncthreads();

    float s = b0[t];
#pragma unroll 4
    for (int k = 0; k < D_FEAT; ++k) s += W0[t * D_FEAT + k] * xf[k];
    z[NHID + t] = fmaxf(s, 0.0f);
    __syncthreads();

    float s1 = b1[t];
#pragma unroll 4
    for (int k = 0; k < 2 * NHID; ++k) s1 += W1[t * (2 * NHID) + k] * z[k];
    h2[t] = fmaxf(s1, 0.0f);
    __syncthreads();

    if (t == 0) {
        float l0 = b2[0], l1 = b2[1];
        for (int k = 0; k < NHID; ++k) {
            l0 += W2[k] * h2[k];
            l1 += W2[NHID + k] * h2[k];
        }
        float m   = fmaxf(l0, l1);
        float lse = m + logf(expf(l0 - m) + expf(l1 - m));
        out[g * 2 + 0] = l0 - lse;
        out[g * 2 + 1] = l1 - lse;
    }
}

extern "C" void kernel_launch(void* const* d_in, const int* in_sizes, int n_in,
                              void* d_out, int out_size, void* d_ws, size_t ws_size,
                              hipStream_t stream) {
    (void)in_sizes; (void)n_in; (void)out_size; (void)ws_size;
    const float* x     = (const float*)d_in[0];
    const int*   ei    = (const int*)  d_in[1];   // [2, E]: src row then dst row
    const int*   batch = (const int*)  d_in[2];
    const float* Wl    = (const float*)d_in[4];
    const float* bl    = (const float*)d_in[5];
    const float* Wr    = (const float*)d_in[6];
    const float* W0    = (const float*)d_in[7];
    const float* b0    = (const float*)d_in[8];
    const float* W1    = (const float*)d_in[9];
    const float* b1    = (const float*)d_in[10];
    const float* W2    = (const float*)d_in[11];
    const float* b2    = (const float*)d_in[12];
    float* out = (float*)d_out;

    // workspace layout
    float* msg    = (float*)d_ws;                         // N * D
    float* deg    = msg + (size_t)N_NODES * D_FEAT;       // N
    float* h      = deg + N_NODES;                        // N * H
    float* pooled = h + (size_t)N_NODES * NHID;           // G * H
    int*   first  = (int*)(pooled + N_GRAPHS * NHID);     // G

    {
        int total = N_NODES * D_FEAT;
        gnn_init<<<(total + 255) / 256, 256, 0, stream>>>(msg, deg, pooled);
    }
    {
        long nth = (long)N_EDGES * 32;                    // one wave per edge
        gnn_scatter<<<(int)((nth + 255) / 256), 256, 0, stream>>>(
            x, ei, ei + N_EDGES, msg, deg);
    }
    {
        int nblk = (N_NODES + 15) / 16;
        gnn_sage_gemm<<<nblk, 256, 0, stream>>>(msg, deg, x, Wl, Wr, bl, h);
    }
    {
        long tot = (long)N_NODES * NHID;
        gnn_pool<<<(int)((tot + 255) / 256), 256, 0, stream>>>(h, batch, pooled);
    }
    gnn_first<<<(N_GRAPHS + 63) / 64, 64, 0, stream>>>(batch, first);
    gnn_head<<<N_GRAPHS, 128, 0, stream>>>(x, first, pooled,
                                           W0, b0, W1, b1, W2, b2, out);
}